// xcorr2_36515811950824
// MI455X (gfx1250) — compile-verified
//
#include <hip/hip_runtime.h>
#include <hip/hip_bf16.h>
#include <stdint.h>

// ---------------------------------------------------------------------------
// Instance-norm'ed per-plane correlation, B=64, C=1, H=W=512, fp32.
// score_b = (P/N - m1*m2) / (sqrt(v1+eps)*sqrt(v2+eps)),  N = 512*512
// where S/Q/P are plane-wide sums. Single streaming pass, memory bound
// (134 MB -> ~5.8us at 23.3 TB/s). Data is staged through LDS with CDNA5
// async global->LDS B128 copies (ASYNCcnt), double-buffered per wave.
// ---------------------------------------------------------------------------

#define PLANE   (512 * 512)          // elements per (b,c) plane
#define CHUNKS  16                   // blocks per plane
#define WAVES   8                    // 256 threads = 8 x wave32
#define STRIDE  (WAVES * 32 * 4)     // elements consumed per block-iteration = 1024
#define ITERS   (PLANE / (CHUNKS * STRIDE))   // = 16
#define EPSF    1e-5f

// Async copy: 16 bytes/lane from global (SGPR base + per-lane byte offset,
// GVS mode) into per-lane LDS address. Tracked with ASYNCcnt.
__device__ __forceinline__ void async_ld_b128(const float* __restrict__ sbase,
                                              uint32_t byte_off, void* lds_ptr) {
    // Generic pointer to LDS: low 32 bits are the workgroup-relative LDS offset.
    uint32_t lds = (uint32_t)(uintptr_t)lds_ptr;
    asm volatile("global_load_async_to_lds_b128 %0, %1, %2"
                 :
                 : "v"(lds), "v"(byte_off), "s"(sbase)
                 : "memory");
}

__device__ __forceinline__ void wait_async_le2() {
    asm volatile("s_wait_asynccnt 2" ::: "memory");
}
__device__ __forceinline__ void wait_async_0() {
    asm volatile("s_wait_asynccnt 0" ::: "memory");
}

__global__ __launch_bounds__(256) void xcorr_stage1(
    const float* __restrict__ x1, const float* __restrict__ x2,
    float* __restrict__ partial /* [B][CHUNKS][5] */) {

    // [buf][input][wave][lane] : 2*2*8*32*16B = 16 KiB
    __shared__ float4 sh[2][2][WAVES][32];
    __shared__ float red[WAVES][5];

    const int b    = blockIdx.x;
    const int c    = blockIdx.y;
    const int w    = threadIdx.x >> 5;
    const int lane = threadIdx.x & 31;

    const float* p1 = x1 + (size_t)b * PLANE;
    const float* p2 = x2 + (size_t)b * PLANE;

    // element offset of this lane's first float4 within the plane
    const uint32_t e0 = (uint32_t)c * (PLANE / CHUNKS) + (uint32_t)w * 128u +
                        (uint32_t)lane * 4u;

    // prologue: stage tile 0 into buffer 0
    async_ld_b128(p1, e0 * 4u, &sh[0][0][w][lane]);
    async_ld_b128(p2, e0 * 4u, &sh[0][1][w][lane]);

    float s1 = 0.f, s2 = 0.f, q1 = 0.f, q2 = 0.f, pp = 0.f;

#define ACC(ax, bx)                                                            \
    do {                                                                       \
        s1 += (ax);                                                            \
        s2 += (bx);                                                            \
        q1 = fmaf((ax), (ax), q1);                                             \
        q2 = fmaf((bx), (bx), q2);                                             \
        pp = fmaf((ax), (bx), pp);                                             \
    } while (0)

#pragma unroll
    for (int t = 0; t < ITERS; ++t) {
        const int buf = t & 1;
        if (t + 1 < ITERS) {
            const uint32_t eb = (e0 + (uint32_t)(t + 1) * STRIDE) * 4u;
            async_ld_b128(p1, eb, &sh[buf ^ 1][0][w][lane]);
            async_ld_b128(p2, eb, &sh[buf ^ 1][1][w][lane]);
            wait_async_le2();   // the 2 just-issued remain in flight
        } else {
            wait_async_0();
        }
        const float4 a  = sh[buf][0][w][lane];
        const float4 bb = sh[buf][1][w][lane];
        ACC(a.x, bb.x);
        ACC(a.y, bb.y);
        ACC(a.z, bb.z);
        ACC(a.w, bb.w);
    }
#undef ACC

    // wave32 tree reduction
#pragma unroll
    for (int off = 16; off > 0; off >>= 1) {
        s1 += __shfl_down(s1, off);
        s2 += __shfl_down(s2, off);
        q1 += __shfl_down(q1, off);
        q2 += __shfl_down(q2, off);
        pp += __shfl_down(pp, off);
    }
    if (lane == 0) {
        red[w][0] = s1; red[w][1] = s2; red[w][2] = q1; red[w][3] = q2; red[w][4] = pp;
    }
    __syncthreads();
    if (threadIdx.x == 0) {
        float t0 = 0.f, t1 = 0.f, t2 = 0.f, t3 = 0.f, t4 = 0.f;
#pragma unroll
        for (int i = 0; i < WAVES; ++i) {
            t0 += red[i][0]; t1 += red[i][1]; t2 += red[i][2];
            t3 += red[i][3]; t4 += red[i][4];
        }
        float* o = partial + ((size_t)b * CHUNKS + c) * 5;
        o[0] = t0; o[1] = t1; o[2] = t2; o[3] = t3; o[4] = t4;
    }
}

__global__ __launch_bounds__(32) void xcorr_stage2(
    const float* __restrict__ partial, float* __restrict__ out) {
    const int b = blockIdx.x;
    const int t = threadIdx.x;

    float s1 = 0.f, s2 = 0.f, q1 = 0.f, q2 = 0.f, pp = 0.f;
    if (t < CHUNKS) {
        const float* q = partial + ((size_t)b * CHUNKS + t) * 5;
        s1 = q[0]; s2 = q[1]; q1 = q[2]; q2 = q[3]; pp = q[4];
    }
#pragma unroll
    for (int off = 16; off > 0; off >>= 1) {
        s1 += __shfl_down(s1, off);
        s2 += __shfl_down(s2, off);
        q1 += __shfl_down(q1, off);
        q2 += __shfl_down(q2, off);
        pp += __shfl_down(pp, off);
    }
    if (t == 0) {
        const float invN = 1.0f / (float)PLANE;
        const float m1 = s1 * invN;
        const float m2 = s2 * invN;
        const float v1 = fmaf(-m1, m1, q1 * invN);
        const float v2 = fmaf(-m2, m2, q2 * invN);
        const float cov = fmaf(-m1, m2, pp * invN);
        out[b] = cov / (sqrtf(v1 + EPSF) * sqrtf(v2 + EPSF));
    }
}

extern "C" void kernel_launch(void* const* d_in, const int* in_sizes, int n_in,
                              void* d_out, int out_size, void* d_ws, size_t ws_size,
                              hipStream_t stream) {
    (void)n_in; (void)out_size; (void)ws_size;
    const float* x1 = (const float*)d_in[0];
    const float* x2 = (const float*)d_in[1];
    float* out     = (float*)d_out;
    float* partial = (float*)d_ws;            // B * CHUNKS * 5 floats (20 KiB)

    const int B = in_sizes[0] / PLANE;        // 64

    dim3 g1((unsigned)B, CHUNKS);
    xcorr_stage1<<<g1, 256, 0, stream>>>(x1, x2, partial);
    xcorr_stage2<<<dim3((unsigned)B), 32, 0, stream>>>(partial, out);
}